// QDuelingGraphNetwork_87058987089991
// MI455X (gfx1250) — compile-verified
//
#include <hip/hip_runtime.h>
#include <hip/hip_bf16.h>

#define N_NODES  4096
#define F        64
#define GCN_FLAT (N_NODES * F)         // 262144
#define N_OBS    (GCN_FLAT + 256)      // 262400
#define HID      1024
#define N_ACT    64
#define N_EDGES  65536
#define KCHUNK   512

typedef __attribute__((ext_vector_type(2))) float v2f;
typedef __attribute__((ext_vector_type(8))) float v8f;

// ---------------------------------------------------------------------------
// Init: deg=1 (self loops), xin[0:GCN_FLAT]=gcn_b[f] (bias folded into scatter
// target), xin tail = flat_obs tail, y1 accumulator = b1.
// ---------------------------------------------------------------------------
__global__ void k_init(const float* __restrict__ gcn_b,
                       const float* __restrict__ flat_obs,
                       const float* __restrict__ b1,
                       float* __restrict__ deg,
                       float* __restrict__ xin,
                       float* __restrict__ y1) {
    int i = blockIdx.x * 256 + threadIdx.x;
    if (i < N_NODES) deg[i] = 1.0f;
    if (i < GCN_FLAT)      xin[i] = gcn_b[i & (F - 1)];
    else if (i < N_OBS)    xin[i] = flat_obs[i];
    if (i < HID) y1[i] = b1[i];
}

__global__ void k_degree(const long long* __restrict__ ei, float* __restrict__ deg) {
    int e = blockIdx.x * 256 + threadIdx.x;
    if (e < N_EDGES) atomicAdd(&deg[(int)ei[N_EDGES + e]], 1.0f);
}

__global__ void k_dinv(const float* __restrict__ deg, float* __restrict__ dinv) {
    int i = blockIdx.x * 256 + threadIdx.x;
    if (i < N_NODES) dinv[i] = rsqrtf(deg[i]);
}

// ---------------------------------------------------------------------------
// GCN dense GEMM: H = X(4096x64) * W(64x64), fp32 WMMA 16x16x4.
// One wave per 16x16 output tile; K looped 0..63 step 4.
// A layout (32-bit 16x4): lanes 0-15 -> M=lane, VGPR0=K0,VGPR1=K1;
//                         lanes 16-31 -> K2,K3.   B mirrors with N across lanes.
// C/D layout: VGPR i, lanes 0-15 -> M=i,N=lane; lanes 16-31 -> M=i+8.
// ---------------------------------------------------------------------------
__global__ void k_gcn_gemm(const float* __restrict__ X,
                           const float* __restrict__ W,
                           float* __restrict__ Hout) {
    int wave = (blockIdx.x * blockDim.x + threadIdx.x) >> 5;  // 1024 tiles
    int lane = threadIdx.x & 31;
    int mt = wave >> 2;          // 256 M-tiles
    int nt = wave & 3;           // 4 N-tiles
    int r  = lane & 15;
    int hi = lane >> 4;

    const float* xrow = X + (mt * 16 + r) * F;
    int ncol = nt * 16 + r;

    v8f c = {};
#pragma unroll
    for (int k0 = 0; k0 < F; k0 += 4) {
        v2f a, b;
        a.x = xrow[k0 + hi * 2];
        a.y = xrow[k0 + hi * 2 + 1];
        b.x = W[(k0 + hi * 2) * F + ncol];
        b.y = W[(k0 + hi * 2 + 1) * F + ncol];
        c = __builtin_amdgcn_wmma_f32_16x16x4_f32(
            /*neg_a=*/false, a, /*neg_b=*/false, b,
            /*c_mod=*/(short)0, c, /*reuse_a=*/false, /*reuse_b=*/false);
    }
#pragma unroll
    for (int i = 0; i < 8; ++i)
        Hout[(mt * 16 + hi * 8 + i) * F + nt * 16 + r] = c[i];
}

// ---------------------------------------------------------------------------
// Edge scatter: xin[d*64+f] += h[s*64+f] * dinv[s]*dinv[d]  (+ self loops)
// 64 consecutive threads share one edge -> fully coalesced h read / atomic add.
// ---------------------------------------------------------------------------
__global__ void k_scatter(const long long* __restrict__ ei,
                          const float* __restrict__ hmat,
                          const float* __restrict__ dinv,
                          float* __restrict__ xin) {
    long idx = (long)blockIdx.x * 256 + threadIdx.x;
    const long total = (long)(N_EDGES + N_NODES) * F;
    if (idx >= total) return;
    int f = (int)(idx & (F - 1));
    int e = (int)(idx >> 6);
    int s, d;
    if (e < N_EDGES) { s = (int)ei[e]; d = (int)ei[N_EDGES + e]; }
    else             { s = d = e - N_EDGES; }
    float norm = dinv[s] * dinv[d];
    atomicAdd(&xin[d * F + f], hmat[s * F + f] * norm);
}

// ---------------------------------------------------------------------------
// Big GEMV: y1 += x[k0:k0+512] @ w1[k0:k0+512, :].  Streams 1.05 GB of w1 —
// the dominant cost (~45us at 23.3 TB/s). x chunk staged in LDS; coalesced
// 256-col stripes; software prefetch 8 rows ahead; one atomicAdd per thread.
// ---------------------------------------------------------------------------
__global__ void k_gemv1(const float* __restrict__ xin,
                        const float* __restrict__ w1,
                        float* __restrict__ y1) {
    __shared__ float xs[KCHUNK];
    int col = blockIdx.x * 256 + threadIdx.x;
    int k0  = blockIdx.y * KCHUNK;
    int kmax = N_OBS - k0; if (kmax > KCHUNK) kmax = KCHUNK;
    for (int t = threadIdx.x; t < KCHUNK; t += 256)
        xs[t] = (k0 + t < N_OBS) ? xin[k0 + t] : 0.0f;
    __syncthreads();

    const float* wp = w1 + (size_t)k0 * HID + col;
    float sum = 0.0f;
    for (int i = 0; i < kmax; ++i) {
        __builtin_prefetch(wp + (size_t)(i + 8) * HID, 0, 1);
        sum = fmaf(xs[i], wp[(size_t)i * HID], sum);
    }
    atomicAdd(&y1[col], sum);
}

// ---------------------------------------------------------------------------
// y2 = relu(relu(y1) @ w2 + b2)  — 4 MB of w2, single pass, relu(y1) in LDS.
// ---------------------------------------------------------------------------
__global__ void k_gemv2(const float* __restrict__ y1,
                        const float* __restrict__ w2,
                        const float* __restrict__ b2,
                        float* __restrict__ y2) {
    __shared__ float ys[HID];
    for (int t = threadIdx.x; t < HID; t += 256) ys[t] = fmaxf(y1[t], 0.0f);
    __syncthreads();
    int col = blockIdx.x * 256 + threadIdx.x;
    float sum = b2[col];
    for (int k = 0; k < HID; ++k)
        sum = fmaf(ys[k], w2[(size_t)k * HID + col], sum);
    y2[col] = fmaxf(sum, 0.0f);
}

// ---------------------------------------------------------------------------
// Dueling heads: out = value + adv - mean(adv).  One 64-thread block.
// ---------------------------------------------------------------------------
__global__ void k_head(const float* __restrict__ y2,
                       const float* __restrict__ wv,
                       const float* __restrict__ bv,
                       const float* __restrict__ wa,
                       const float* __restrict__ ba,
                       float* __restrict__ out) {
    __shared__ float sadv[N_ACT];
    __shared__ float sval;
    int j = threadIdx.x;
    float adv = ba[j];
    for (int k = 0; k < HID; ++k) adv = fmaf(y2[k], wa[k * N_ACT + j], adv);
    sadv[j] = adv;
    if (j == 0) {
        float v = bv[0];
        for (int k = 0; k < HID; ++k) v = fmaf(y2[k], wv[k], v);
        sval = v;
    }
    __syncthreads();
    float m = 0.0f;
    for (int t = 0; t < N_ACT; ++t) m += sadv[t];
    m *= (1.0f / N_ACT);
    out[j] = sval + sadv[j] - m;
}

extern "C" void kernel_launch(void* const* d_in, const int* in_sizes, int n_in,
                              void* d_out, int out_size, void* d_ws, size_t ws_size,
                              hipStream_t stream) {
    const float*     node_x   = (const float*)d_in[0];
    const float*     flat_obs = (const float*)d_in[1];
    const float*     gcn_w    = (const float*)d_in[2];
    const float*     gcn_b    = (const float*)d_in[3];
    const float*     w1       = (const float*)d_in[4];
    const float*     b1       = (const float*)d_in[5];
    const float*     w2       = (const float*)d_in[6];
    const float*     b2       = (const float*)d_in[7];
    const float*     wv       = (const float*)d_in[8];
    const float*     bv       = (const float*)d_in[9];
    const float*     wa       = (const float*)d_in[10];
    const float*     ba       = (const float*)d_in[11];
    const long long* ei       = (const long long*)d_in[12];  // int64 [2, N_EDGES]

    float* ws   = (float*)d_ws;
    float* deg  = ws;                 // 4096
    float* dinv = ws + 4096;          // 4096
    float* hbuf = ws + 8192;          // 262144
    float* xin  = hbuf + GCN_FLAT;    // 262400
    float* y1   = xin + N_OBS;        // 1024
    float* y2   = y1 + HID;           // 1024
    float* out  = (float*)d_out;      // 64

    k_init  <<<(N_OBS + 255) / 256,   256, 0, stream>>>(gcn_b, flat_obs, b1, deg, xin, y1);
    k_degree<<<(N_EDGES + 255) / 256, 256, 0, stream>>>(ei, deg);
    k_dinv  <<<(N_NODES + 255) / 256, 256, 0, stream>>>(deg, dinv);

    // 1024 16x16 tiles, 8 waves (tiles) per 256-thread block -> 128 blocks
    k_gcn_gemm<<<128, 256, 0, stream>>>(node_x, gcn_w, hbuf);

    long total = (long)(N_EDGES + N_NODES) * F;
    k_scatter<<<(int)((total + 255) / 256), 256, 0, stream>>>(ei, hbuf, dinv, xin);

    dim3 g1(HID / 256, (N_OBS + KCHUNK - 1) / KCHUNK);   // (4, 513) split-K
    k_gemv1<<<g1, 256, 0, stream>>>(xin, w1, y1);

    k_gemv2<<<HID / 256, 256, 0, stream>>>(y1, w2, b2, y2);
    k_head <<<1, N_ACT, 0, stream>>>(y2, wv, bv, wa, ba, out);
}